// RBF_39264591020613
// MI455X (gfx1250) — compile-verified
//
#include <hip/hip_runtime.h>
#include <hip/hip_bf16.h>

typedef __attribute__((ext_vector_type(2))) float v2f;
typedef __attribute__((ext_vector_type(8))) float v8f;

__device__ __forceinline__ float softplus_f(float v) {
    // inputs are in [0,1]; plain log(1+exp(x)) is numerically fine there
    return logf(1.0f + __expf(v));
}

// One wave (32 lanes) computes one 16x16 tile of cov via 4 chained
// V_WMMA_F32_16X16X4_F32 (K = 16 total, full fp32).
// cov[i,j] = variance * exp( a_i . b_j - 0.5*||a_i||^2 - 0.5*||b_j||^2 ),
// a = x/scale, b = xx/scale.
__global__ __launch_bounds__(256) void RBF_39264591020613_kernel(
    const float* __restrict__ x,            // [N,16]
    const float* __restrict__ xx,           // [M,16]
    const float* __restrict__ scale_ff,     // [16]
    const float* __restrict__ variance_ff,  // [1]
    float* __restrict__ out,                // [N,M]
    int N, int M)
{
    const int lane = threadIdx.x & 31;
    const int wave = threadIdx.x >> 5;
    const int n    = lane & 15;   // tile-local column (and A-row id for loads)
    const int h    = lane >> 4;   // half-wave select
    const int i0   = blockIdx.y * 128 + wave * 16;  // output row tile base
    const int j0   = blockIdx.x * 16;               // output col tile base

    // Inverse length-scales for the 8 feature dims this lane touches:
    // d = 4t + 2h + {0,1}, t = 0..3
    float is[8];
#pragma unroll
    for (int t = 0; t < 4; ++t) {
        const int d0 = 4 * t + 2 * h;
        is[2 * t]     = 1.0f / softplus_f(scale_ff[d0]);
        is[2 * t + 1] = 1.0f / softplus_f(scale_ff[d0 + 1]);
    }

    const float* __restrict__ xr = x  + (size_t)(i0 + n) * 16;  // A row (i0+n)
    const float* __restrict__ br = xx + (size_t)(j0 + n) * 16;  // B col (j0+n)

    v8f acc = {};
    float na_p = 0.0f;  // partial ||a_row||^2 over this lane's 8 dims
    float nb_p = 0.0f;  // partial ||b_col||^2 over this lane's 8 dims
#pragma unroll
    for (int t = 0; t < 4; ++t) {
        const int d0 = 4 * t + 2 * h;
        v2f a, b;
        a.x = xr[d0]     * is[2 * t];
        a.y = xr[d0 + 1] * is[2 * t + 1];
        b.x = br[d0]     * is[2 * t];
        b.y = br[d0 + 1] * is[2 * t + 1];
        na_p = fmaf(a.x, a.x, fmaf(a.y, a.y, na_p));
        nb_p = fmaf(b.x, b.x, fmaf(b.y, b.y, nb_p));
        // D = A(16x4,f32) * B(4x16,f32) + C ; K-slice t covers dims 4t..4t+3
        acc = __builtin_amdgcn_wmma_f32_16x16x4_f32(
            /*neg_a=*/false, a, /*neg_b=*/false, b,
            /*c_mod=*/(short)0, acc, /*reuse_a=*/false, /*reuse_b=*/false);
    }

    // Lanes n and n^16 together hold all 16 dims of row/col n -> one xor-shuffle
    const float na_full = na_p + __shfl_xor(na_p, 16, 32);  // ||a_{i0+n}||^2
    const float nb_full = nb_p + __shfl_xor(nb_p, 16, 32);  // ||b_{j0+n}||^2

    const float variance = softplus_f(variance_ff[0]);

#pragma unroll
    for (int v = 0; v < 8; ++v) {
        const int r = v + 8 * h;                         // tile-local row of acc[v]
        const float na_r = __shfl(na_full, r, 32);       // broadcast row norm
        const float e = acc[v] - 0.5f * (na_r + nb_full);
        out[(size_t)(i0 + r) * (size_t)M + (size_t)(j0 + n)] = variance * __expf(e);
    }
}

extern "C" void kernel_launch(void* const* d_in, const int* in_sizes, int n_in,
                              void* d_out, int out_size, void* d_ws, size_t ws_size,
                              hipStream_t stream) {
    (void)n_in; (void)d_ws; (void)ws_size; (void)out_size;
    const float* x           = (const float*)d_in[0];
    const float* xx          = (const float*)d_in[1];
    const float* scale_ff    = (const float*)d_in[2];
    const float* variance_ff = (const float*)d_in[3];
    float* out = (float*)d_out;

    const int D = 16;
    const int N = in_sizes[0] / D;  // 4096
    const int M = in_sizes[1] / D;  // 4096

    dim3 grid(M / 16, N / 128);     // one wave per 16x16 tile, 8 waves/block
    RBF_39264591020613_kernel<<<grid, 256, 0, stream>>>(
        x, xx, scale_ff, variance_ff, out, N, M);
}